// AssociativeScanGLRU_75350906241525
// MI455X (gfx1250) — compile-verified
//
#include <hip/hip_runtime.h>

// GLRU associative scan, MI455X (gfx1250, wave32).
// Memory-bound: 201 MB read + 67 MB write => ~11.5 us floor @ 23.3 TB/s.
// Phase 1: chunk aggregates (direct clause'd b128 loads, warms L2).
// Phase 2: tiny spine scan over 64 chunk aggregates per lane.
// Phase 3: async-to-LDS pipelined apply pass (gfx1250 GLOBAL_LOAD_ASYNC_TO_LDS,
//          ASYNCcnt-tracked, VGPR-free in-flight traffic), NT stores for y.
// No WMMA: the recurrence is elementwise per channel (no contraction dim).

typedef float v4f __attribute__((ext_vector_type(4)));

#define DH     1024
#define BATCH  4
#define TLEN   4096
#define NCH    64              // time chunks
#define CLEN   (TLEN / NCH)    // 64 steps per chunk
#define COLS   (DH / 4)        // 256 float4 columns per batch
#define XSTR   (3 * DH)        // x row stride in elements
#define NSTAGE 8               // async pipeline depth (stages of 1 timestep)

// ---- fast scalar math (gfx1250 TRANS ops) ----
static __device__ __forceinline__ float fexp2(float x) {
#if __has_builtin(__builtin_amdgcn_exp2f)
  return __builtin_amdgcn_exp2f(x);
#else
  return exp2f(x);
#endif
}
static __device__ __forceinline__ float frcp(float x) {
#if __has_builtin(__builtin_amdgcn_rcpf)
  return __builtin_amdgcn_rcpf(x);
#else
  return 1.0f / x;
#endif
}
static __device__ __forceinline__ float fsigmoid(float x) {
  return frcp(1.0f + fexp2(-1.4426950408889634f * x));
}
static __device__ __forceinline__ float ftanh(float x) {
#if __has_builtin(__builtin_amdgcn_tanhf)
  return __builtin_amdgcn_tanhf(x);          // gfx1250 v_tanh_f32
#elif __has_builtin(__builtin_amdgcn_tanh_f32)
  return __builtin_amdgcn_tanh_f32(x);
#else
  return 2.0f * frcp(1.0f + fexp2(-2.8853900817779268f * x)) - 1.0f;
#endif
}
static __device__ __forceinline__ v4f vtanh(v4f a) {
  v4f r; r.x = ftanh(a.x); r.y = ftanh(a.y); r.z = ftanh(a.z); r.w = ftanh(a.w); return r;
}
static __device__ __forceinline__ v4f vsig(v4f a) {
  v4f r; r.x = fsigmoid(a.x); r.y = fsigmoid(a.y); r.z = fsigmoid(a.z); r.w = fsigmoid(a.w); return r;
}

// ---- gfx1250 async / wait primitives (guarded builtin -> asm fallback) ----
template <int N>
static __device__ __forceinline__ void wait_async() {
#if __has_builtin(__builtin_amdgcn_s_wait_asynccnt)
  __builtin_amdgcn_s_wait_asynccnt(N);
#else
  asm volatile("s_wait_asynccnt %0" :: "n"(N) : "memory");
#endif
}
static __device__ __forceinline__ void wait_ds0() {
#if __has_builtin(__builtin_amdgcn_s_wait_dscnt)
  __builtin_amdgcn_s_wait_dscnt(0);
#else
  asm volatile("s_wait_dscnt 0x0" ::: "memory");
#endif
}
// GVS form: mem = SGPR64(base) + VGPR32(byte off); dest = per-lane LDS byte addr.
static __device__ __forceinline__ void async_ld_b128(unsigned lds_off, unsigned voff,
                                                     const float* sbase) {
  asm volatile("global_load_async_to_lds_b128 %0, %1, %2"
               :: "v"(lds_off), "v"(voff), "s"(sbase)
               : "memory");
}

// ---- Phase 1: per-chunk aggregates (a = prod f, s = local scan tail) ----
// Carry folds in exactly by initializing s = carry on chunk 0.
__global__ void __launch_bounds__(COLS) glru_agg(
    const float* __restrict__ x, const float* __restrict__ carry,
    float* __restrict__ aggA, float* __restrict__ aggS) {
  const int b  = blockIdx.x / NCH;
  const int ch = blockIdx.x % NCH;
  const int d0 = threadIdx.x * 4;
  const float* px = x + (size_t)(b * TLEN + ch * CLEN) * XSTR + d0;

  v4f a = {1.0f, 1.0f, 1.0f, 1.0f};
  v4f s = {0.0f, 0.0f, 0.0f, 0.0f};
  if (ch == 0) s = *(const v4f*)(carry + b * DH + d0);

#pragma unroll 8
  for (int t = 0; t < CLEN; ++t) {
    v4f xi = *(const v4f*)(px + t * XSTR);        // pre-activation input
    v4f xg = *(const v4f*)(px + t * XSTR + DH);   // pre-activation input gate
    v4f ig = vsig(xg);
    v4f f  = 1.0f - ig;                           // forget gate
    s = f * s + vtanh(xi) * ig;
    a = a * f;
  }
  const int o = (b * NCH + ch) * DH + d0;
  *(v4f*)(aggA + o) = a;
  *(v4f*)(aggS + o) = s;
}

// ---- Phase 2: scan the chunk aggregates per lane (tiny) ----
__global__ void __launch_bounds__(256) glru_spine(
    const float* __restrict__ aggA, const float* __restrict__ aggS,
    float* __restrict__ pre, float* __restrict__ hlast) {
  const int tid = blockIdx.x * 256 + threadIdx.x;   // 0..4095 = b*DH + d
  const int b = tid / DH, d = tid % DH;
  float h = 0.0f;
#pragma unroll
  for (int k = 0; k < NCH; ++k) {
    const int idx = (b * NCH + k) * DH + d;
    pre[idx] = h;                       // h entering chunk k
    h = fmaf(aggA[idx], h, aggS[idx]);  // h leaving chunk k
  }
  hlast[tid] = h;                       // d_out[0 .. 4095]
}

// ---- Phase 3: async-to-LDS pipelined apply pass ----
// Per-thread LDS FIFO: NSTAGE stages x 3 streams x b128. Each thread consumes
// exactly what it fetched -> no cross-thread sync, only asynccnt/dscnt waits.
__global__ void __launch_bounds__(COLS) glru_apply(
    const float* __restrict__ x, const float* __restrict__ carry,
    const float* __restrict__ pre, float* __restrict__ y) {
  extern __shared__ v4f smem[];         // NSTAGE * 3 * COLS v4f = 96 KB
  const int b   = blockIdx.x / NCH;
  const int ch  = blockIdx.x % NCH;
  const int tid = threadIdx.x;
  const int d0  = tid * 4;
  const float* px = x + (size_t)(b * TLEN + ch * CLEN) * XSTR;  // block-uniform base
  float* py = y + (size_t)(b * TLEN + ch * CLEN) * DH + d0;

  // LDS byte offset of this thread's slot (low 32 bits of flat addr == LDS offset)
  const unsigned lds_base = (unsigned)(uintptr_t)(&smem[0]);
#define LDSOFF(slot, k) (lds_base + (unsigned)((((slot) * 3 + (k)) * COLS + tid) * 16))
#define GVOFF(t, k)     ((unsigned)(((t) * XSTR + (k) * DH + d0) * 4))

  // prologue: fill all NSTAGE stages (3*NSTAGE async b128 copies in flight)
#pragma unroll
  for (int s = 0; s < NSTAGE; ++s) {
    async_ld_b128(LDSOFF(s, 0), GVOFF(s, 0), px);
    async_ld_b128(LDSOFF(s, 1), GVOFF(s, 1), px);
    async_ld_b128(LDSOFF(s, 2), GVOFF(s, 2), px);
  }

  v4f h = *(const v4f*)(pre + (b * NCH + ch) * DH + d0);
  if (ch == 0) h += *(const v4f*)(carry + b * DH + d0);   // pre[chunk 0] == 0

  // steady state: wait oldest stage, consume, refill
  for (int t = 0; t < CLEN - NSTAGE; ++t) {
    const int slot = t % NSTAGE;
    wait_async<3 * (NSTAGE - 1)>();     // stage t's 3 copies have landed (in-order done)
    v4f xi = smem[(slot * 3 + 0) * COLS + tid];
    v4f xg = smem[(slot * 3 + 1) * COLS + tid];
    v4f xo = smem[(slot * 3 + 2) * COLS + tid];
    v4f ig = vsig(xg);
    v4f f  = 1.0f - ig;
    h = f * h + vtanh(xi) * ig;
    v4f yv = vtanh(h) * vsig(xo);
    __builtin_nontemporal_store(yv, (v4f*)(py + t * DH));
    wait_ds0();                         // slot's ds reads drained -> safe to overwrite
    const int tn = t + NSTAGE;
    async_ld_b128(LDSOFF(slot, 0), GVOFF(tn, 0), px);
    async_ld_b128(LDSOFF(slot, 1), GVOFF(tn, 1), px);
    async_ld_b128(LDSOFF(slot, 2), GVOFF(tn, 2), px);
  }

  // tail: everything already in LDS
  wait_async<0>();
#pragma unroll
  for (int t = CLEN - NSTAGE; t < CLEN; ++t) {
    const int slot = t % NSTAGE;
    v4f xi = smem[(slot * 3 + 0) * COLS + tid];
    v4f xg = smem[(slot * 3 + 1) * COLS + tid];
    v4f xo = smem[(slot * 3 + 2) * COLS + tid];
    v4f ig = vsig(xg);
    v4f f  = 1.0f - ig;
    h = f * h + vtanh(xi) * ig;
    v4f yv = vtanh(h) * vsig(xo);
    __builtin_nontemporal_store(yv, (v4f*)(py + t * DH));
  }
#undef LDSOFF
#undef GVOFF
}

extern "C" void kernel_launch(void* const* d_in, const int* in_sizes, int n_in,
                              void* d_out, int out_size, void* d_ws, size_t ws_size,
                              hipStream_t stream) {
  (void)in_sizes; (void)n_in; (void)out_size; (void)ws_size;
  const float* x     = (const float*)d_in[0];  // (4, 4096, 3072) f32
  const float* carry = (const float*)d_in[1];  // (4, 1024) f32
  float* out = (float*)d_out;                  // [h_last (4*1024) | y (4*4096*1024)]

  // Scratch layout (3 MiB, fully written before read each call):
  float* aggA = (float*)d_ws;                  // (4, 64, 1024)
  float* aggS = aggA + BATCH * NCH * DH;       // (4, 64, 1024)
  float* pre  = aggS + BATCH * NCH * DH;       // (4, 64, 1024) chunk-entry h

  const size_t smem_bytes = (size_t)NSTAGE * 3 * COLS * sizeof(v4f);  // 96 KB

  glru_agg  <<<BATCH * NCH, COLS, 0, stream>>>(x, carry, aggA, aggS);
  glru_spine<<<(BATCH * DH) / 256, 256, 0, stream>>>(aggA, aggS, pre, out);
  glru_apply<<<BATCH * NCH, COLS, smem_bytes, stream>>>(x, carry, pre, out + BATCH * DH);
}